// ContrastiveLoss_3246995276360
// MI455X (gfx1250) — compile-verified
//
#include <hip/hip_runtime.h>
#include <hip/hip_bf16.h>

typedef __attribute__((ext_vector_type(2))) float v2f;
typedef __attribute__((ext_vector_type(8))) float v8f;

#define MARGIN 0.5f
#define EPSC   1e-8f

// One wave processes 16 pairs per iteration using V_WMMA_F32_16X16X4_F32.
// A = 16 src rows (16x64), B-columns = 16 tgt rows. Accumulating 16 K=4 chunks:
//   diag(A x B) = pair dots, diag(A x A) = |a|^2, diag(B x B) = |b|^2
// (f32 A- and B-operand lane/VGPR layouts coincide, so the same registers feed both sides).
__global__ void pair_loss_kernel(const float* __restrict__ embS,
                                 const float* __restrict__ embT,
                                 const int*   __restrict__ srcIdx,
                                 const int*   __restrict__ tgtIdx,
                                 int nPairs, float weight, int isPos,
                                 float* __restrict__ partials) {
    const int lane   = threadIdx.x & 31;
    const int waveId = threadIdx.x >> 5;
    const int wavesPerBlock = blockDim.x >> 5;
    const int globalWave = blockIdx.x * wavesPerBlock + waveId;
    const int numWaves   = gridDim.x * wavesPerBlock;

    const int p  = lane & 15;   // pair slot within group (M and N index)
    const int hh = lane >> 4;   // lane half selects K sub-pair within a chunk

    const int numGroups = (nPairs + 15) >> 4;

    float waveAcc = 0.0f;

    for (int g = globalWave; g < numGroups; g += numWaves) {
        const int pg = g * 16 + p;
        const int pi = pg < nPairs ? pg : (nPairs - 1);   // clamp: keep EXEC all-1s

        const size_t ia = (size_t)(unsigned)srcIdx[pi];
        const size_t ib = (size_t)(unsigned)tgtIdx[pi];
        const float* arow = embS + ia * 64;
        const float* brow = embT + ib * 64;

        // Gather: per chunk c, lane needs row[4c + 2*hh], row[4c + 2*hh + 1]
        // -> one b64 load; lanes p and p+16 cover contiguous 16B of the row.
        v2f A[16], B[16];
#pragma unroll
        for (int c = 0; c < 16; ++c) {
            A[c] = *reinterpret_cast<const v2f*>(arow + 4 * c + 2 * hh);
            B[c] = *reinterpret_cast<const v2f*>(brow + 4 * c + 2 * hh);
        }

        v8f Dab = {}; v8f Daa = {}; v8f Dbb = {};
#pragma unroll
        for (int c = 0; c < 16; ++c) {
            // three independent accumulation chains -> latency hiding
            Dab = __builtin_amdgcn_wmma_f32_16x16x4_f32(false, A[c], false, B[c],
                                                        (short)0, Dab, false, false);
            Daa = __builtin_amdgcn_wmma_f32_16x16x4_f32(false, A[c], false, A[c],
                                                        (short)0, Daa, false, false);
            Dbb = __builtin_amdgcn_wmma_f32_16x16x4_f32(false, B[c], false, B[c],
                                                        (short)0, Dbb, false, false);
        }

        // Diagonal extraction. C/D layout: VGPR r, lanes 0-15 -> (M=r, N=lane);
        // lanes 16-31 -> (M=r+8, N=lane-16). So diag m<8 sits at lane m in acc[m],
        // diag m>=8 sits at lane m+16 in acc[m-8].
        float dab = 0.0f, daa = 0.0f, dbb = 0.0f;
#pragma unroll
        for (int r = 0; r < 8; ++r) {
            const bool sel = (lane == r) || (lane == r + 24);
            dab = sel ? Dab[r] : dab;
            daa = sel ? Daa[r] : daa;
            dbb = sel ? Dbb[r] : dbb;
        }
        // Move diags 8..15 (lanes 24..31) down to lanes 8..15.
        const int srcLane = (lane & 8) ? (lane + 16) : lane;
        dab = __shfl(dab, srcLane, 32);
        daa = __shfl(daa, srcLane, 32);
        dbb = __shfl(dbb, srcLane, 32);

        const float denom = fmaxf(sqrtf(daa) * sqrtf(dbb), EPSC);
        const float cosv  = dab / denom;
        float loss = isPos ? (1.0f - cosv) : fmaxf(cosv - MARGIN, 0.0f);
        if (hh != 0 || pg >= nPairs) loss = 0.0f;  // only lanes 0..15 carry valid pairs
        waveAcc += loss;
    }

    waveAcc *= weight;

    // Wave reduction (wave32)
#pragma unroll
    for (int off = 16; off > 0; off >>= 1)
        waveAcc += __shfl_xor(waveAcc, off, 32);

    __shared__ float sm[32];
    if (lane == 0) sm[waveId] = waveAcc;
    __syncthreads();

    if (waveId == 0) {
        float v = (lane < wavesPerBlock) ? sm[lane] : 0.0f;
#pragma unroll
        for (int off = 4; off > 0; off >>= 1)
            v += __shfl_xor(v, off, 32);
        if (lane == 0) partials[blockIdx.x] = v;
    }
}

__global__ void reduce_kernel(const float* __restrict__ partials, int n,
                              float invCount, float* __restrict__ out) {
    __shared__ float s[256];
    float acc = 0.0f;
    for (int i = threadIdx.x; i < n; i += 256) acc += partials[i];
    s[threadIdx.x] = acc;
    __syncthreads();
#pragma unroll
    for (int st = 128; st > 0; st >>= 1) {
        if ((int)threadIdx.x < st) s[threadIdx.x] += s[threadIdx.x + st];
        __syncthreads();
    }
    if (threadIdx.x == 0) out[0] = s[0] * invCount;
}

extern "C" void kernel_launch(void* const* d_in, const int* in_sizes, int n_in,
                              void* d_out, int out_size, void* d_ws, size_t ws_size,
                              hipStream_t stream) {
    const float* embU = (const float*)d_in[0];
    const float* embI = (const float*)d_in[1];
    const float* embG = (const float*)d_in[2];
    const int* pus = (const int*)d_in[3];
    const int* put = (const int*)d_in[4];
    const int* pgs = (const int*)d_in[5];
    const int* pgt = (const int*)d_in[6];
    const int* nus = (const int*)d_in[7];
    const int* nut = (const int*)d_in[8];
    const int* ngs = (const int*)d_in[9];
    const int* ngt = (const int*)d_in[10];

    const int nPU = in_sizes[3];
    const int nPG = in_sizes[5];
    const int nNU = in_sizes[7];
    const int nNG = in_sizes[9];

    constexpr int BLOCKS  = 1024;
    constexpr int THREADS = 256;   // 8 wave32 per block
    float* partials = (float*)d_ws; // 4 * BLOCKS floats = 16 KB

    pair_loss_kernel<<<BLOCKS, THREADS, 0, stream>>>(embU, embI, pus, put, nPU, 1.0f, 1,
                                                     partials + 0 * BLOCKS);
    pair_loss_kernel<<<BLOCKS, THREADS, 0, stream>>>(embG, embI, pgs, pgt, nPG, 2.0f, 1,
                                                     partials + 1 * BLOCKS);
    pair_loss_kernel<<<BLOCKS, THREADS, 0, stream>>>(embU, embI, nus, nut, nNU, 1.0f, 0,
                                                     partials + 2 * BLOCKS);
    pair_loss_kernel<<<BLOCKS, THREADS, 0, stream>>>(embG, embI, ngs, ngt, nNG, 2.0f, 0,
                                                     partials + 3 * BLOCKS);

    const float invCount = 1.0f / (float)(nPU + nPG + nNU + nNG);
    reduce_kernel<<<1, THREADS, 0, stream>>>(partials, 4 * BLOCKS, invCount, (float*)d_out);
}